// Propagation_12335146074511
// MI455X (gfx1250) — compile-verified
//
#include <hip/hip_runtime.h>
#include <cstdint>

// Problem constants from the reference: B=16384, L=60, C=48.
constexpr int Bn = 16384;
constexpr int Ln = 60;
constexpr int Cn = 48;

typedef float vf4 __attribute__((ext_vector_type(4)));

// Reduce across a 16-lane half-wave (wave32). Masks 1,2,4,8 stay inside each half.
__device__ __forceinline__ float hred16(float v) {
    v += __shfl_xor(v, 1, 32);
    v += __shfl_xor(v, 2, 32);
    v += __shfl_xor(v, 4, 32);
    v += __shfl_xor(v, 8, 32);
    return v;
}

struct F4 { float v[4]; };
__device__ __forceinline__ F4 loadF4(const float4* p) {
    float4 t = *p;
    F4 r; r.v[0] = t.x; r.v[1] = t.y; r.v[2] = t.z; r.v[3] = t.w;
    return r;
}
__device__ __forceinline__ F4 loadF4NT(const float* p) {
    vf4 t = __builtin_nontemporal_load((const vf4*)p);
    F4 r; r.v[0] = t.x; r.v[1] = t.y; r.v[2] = t.z; r.v[3] = t.w;
    return r;
}

// Per wave: 2 batch rows. Per layer the wave consumes 8 tensors * 2b * 192B = 3072B,
// double-buffered in LDS (6144B/wave, 48KB/block of 8 waves; WGP has 320KB).
// Loads use gfx1250 async global->LDS (ASYNCcnt), overlapped 2 layers deep.
// The 1.51 GB coefficient stream has zero reuse -> th:TH_LOAD_NT keeps it out of L2.
__global__ __launch_bounds__(256) void prop_scan_kernel(
    const float* __restrict__ g_td,  const float* __restrict__ g_tdf,
    const float* __restrict__ g_tmd, const float* __restrict__ g_tmdf,
    const float* __restrict__ g_rsd, const float* __restrict__ g_rsdf,
    const float* __restrict__ g_ad,  const float* __restrict__ g_adf,
    const float* __restrict__ g_urt, const float* __restrict__ g_fd,
    const float* __restrict__ g_fdf, float* __restrict__ out)
{
    __shared__ __align__(16) char smem[8 * 2 * 3072];

    const int tid  = threadIdx.x;
    const int wave = tid >> 5;          // 0..7
    const int lane = tid & 31;
    const int half = lane >> 4;         // which of the 2 batch rows (compute map)
    const int sub  = lane & 15;         // float4 slot within the 48-ch row
    const bool act = sub < 12;          // 12 * float4 = 48 channels

    const int b0 = blockIdx.x * 16 + wave * 2;   // this wave's first batch row
    const int bC = b0 + half;                    // this lane's batch row

    // Low 32 bits of a flat LDS address are the LDS byte offset (aperture is in
    // the high half) -> usable directly as the async-load VDST LDS address.
    const uint32_t ldsWave = (uint32_t)(size_t)&smem[wave * 2 * 3072];

    // Async-load lane map (lanes 0..23 active): bh = lane/12 picks the batch row,
    // ach = lane%12 picks the 16B chunk. One instruction per tensor per layer.
    const int abh = lane / 12;
    const int ach = lane - abh * 12;
    const uint32_t voff0 = (uint32_t)(b0 + abh) * (uint32_t)(Ln * Cn * 4)
                         + (uint32_t)ach * 16u;

    // LDS buffer layout per wave-buffer: q*384 + bh*192 + chunk*16 == q*384 + lane*16.
#define ASYNC_BATCH(BUFSEL, LAYER)                                                   \
    if (lane < 24) {                                                                 \
        uint32_t _lb = ldsWave + (uint32_t)(BUFSEL) * 3072u + (uint32_t)lane * 16u;  \
        uint32_t _vo = voff0 + (uint32_t)(LAYER) * (uint32_t)(Cn * 4);               \
        asm volatile("global_load_async_to_lds_b128 %0, %1, %2 th:TH_LOAD_NT" :: "v"(_lb +    0u), "v"(_vo), "s"(g_td)   : "memory"); \
        asm volatile("global_load_async_to_lds_b128 %0, %1, %2 th:TH_LOAD_NT" :: "v"(_lb +  384u), "v"(_vo), "s"(g_tdf)  : "memory"); \
        asm volatile("global_load_async_to_lds_b128 %0, %1, %2 th:TH_LOAD_NT" :: "v"(_lb +  768u), "v"(_vo), "s"(g_tmd)  : "memory"); \
        asm volatile("global_load_async_to_lds_b128 %0, %1, %2 th:TH_LOAD_NT" :: "v"(_lb + 1152u), "v"(_vo), "s"(g_tmdf) : "memory"); \
        asm volatile("global_load_async_to_lds_b128 %0, %1, %2 th:TH_LOAD_NT" :: "v"(_lb + 1536u), "v"(_vo), "s"(g_rsd)  : "memory"); \
        asm volatile("global_load_async_to_lds_b128 %0, %1, %2 th:TH_LOAD_NT" :: "v"(_lb + 1920u), "v"(_vo), "s"(g_rsdf) : "memory"); \
        asm volatile("global_load_async_to_lds_b128 %0, %1, %2 th:TH_LOAD_NT" :: "v"(_lb + 2304u), "v"(_vo), "s"(g_ad)   : "memory"); \
        asm volatile("global_load_async_to_lds_b128 %0, %1, %2 th:TH_LOAD_NT" :: "v"(_lb + 2688u), "v"(_vo), "s"(g_adf)  : "memory"); \
    }

    // Prime the pipeline: layers 0 and 1 in flight while we do the level-0 work.
    ASYNC_BATCH(0, 0)
    ASYNC_BATCH(1, 1)

    // ---- level 0: carries + top-of-atmosphere outputs (global loads, overlap async) ----
    F4 fd  = {{0.f, 0.f, 0.f, 0.f}};
    F4 fdf = {{0.f, 0.f, 0.f, 0.f}};
    float p0 = 0.f, p1 = 0.f, p2 = 0.f;
    if (act) {
        fd  = loadF4NT(g_fd  + bC * Cn + sub * 4);
        fdf = loadF4NT(g_fdf + bC * Cn + sub * 4);
        F4 ur = loadF4NT(g_urt + bC * Cn + sub * 4);
#pragma unroll
        for (int k = 0; k < 4; ++k) {
            p0 += fd.v[k];
            p1 += fdf.v[k];
            p2 += fd.v[k] * ur.v[k];
        }
    }
    p0 = hred16(p0); p1 = hred16(p1); p2 = hred16(p2);

    float* o_dd  = out;                          // down_direct  [B, L+1]
    float* o_ddf = out + Bn * (Ln + 1);          // down_diffuse [B, L+1]
    float* o_up  = out + 2 * Bn * (Ln + 1);      // up_diffuse   [B, L+1]
    float* o_ab  = out + 3 * Bn * (Ln + 1);      // absorbed     [B, L]
    if (sub == 0) {
        __builtin_nontemporal_store(p0, &o_dd [bC * (Ln + 1)]);
        __builtin_nontemporal_store(p1, &o_ddf[bC * (Ln + 1)]);
        __builtin_nontemporal_store(p2, &o_up [bC * (Ln + 1)]);
    }

    const int ci = half * 12 + sub;              // float4 index within a 384B q-slab

    for (int l = 0; l < Ln; ++l) {
        // Current layer's 8 async loads are complete once <=8 (the next layer's)
        // remain outstanding; async loads complete in issue order.
        if (l >= Ln - 1) { asm volatile("s_wait_asynccnt 0x0" ::: "memory"); }
        else             { asm volatile("s_wait_asynccnt 0x8" ::: "memory"); }

        const float4* bufq = (const float4*)(smem + wave * 6144 + (l & 1) * 3072);

        F4 z = {{0.f, 0.f, 0.f, 0.f}};
        F4 ctd = z, ctdf = z, ctmd = z, ctmdf = z, crsd = z, crsdf = z, cad = z, cadf = z;
        if (act) {
            ctd   = loadF4(bufq + 0 * 24 + ci);
            ctdf  = loadF4(bufq + 1 * 24 + ci);
            ctmd  = loadF4(bufq + 2 * 24 + ci);
            ctmdf = loadF4(bufq + 3 * 24 + ci);
            crsd  = loadF4(bufq + 4 * 24 + ci);
            crsdf = loadF4(bufq + 5 * 24 + ci);
            cad   = loadF4(bufq + 6 * 24 + ci);
            cadf  = loadF4(bufq + 7 * 24 + ci);
        }

        float pa = 0.f, pdd = 0.f, pddf = 0.f, pup = 0.f;
#pragma unroll
        for (int k = 0; k < 4; ++k) {
            const float FD = fd.v[k], FDF = fdf.v[k];
            const float ab   = FD * cad.v[k]  + FDF * cadf.v[k];
            const float nfd  = FD * ctd.v[k];
            const float nfdf = FD * ctmd.v[k] + FDF * (ctdf.v[k] + ctmdf.v[k]);
            const float up   = FD * crsd.v[k] + FDF * crsdf.v[k];
            fd.v[k]  = nfd;
            fdf.v[k] = nfdf;
            pa += ab; pdd += nfd; pddf += nfdf; pup += up;
        }
        pa   = hred16(pa);
        pdd  = hred16(pdd);
        pddf = hred16(pddf);
        pup  = hred16(pup);

        if (sub == 0) {
            __builtin_nontemporal_store(pdd,  &o_dd [bC * (Ln + 1) + l + 1]);
            __builtin_nontemporal_store(pddf, &o_ddf[bC * (Ln + 1) + l + 1]);
            __builtin_nontemporal_store(pup,  &o_up [bC * (Ln + 1) + l + 1]);
            __builtin_nontemporal_store(pa,   &o_ab [bC * Ln + l]);
        }

        // Refill the buffer we just consumed with layer l+2. Make sure our LDS
        // reads have drained before the async engine overwrites the buffer.
        if (l + 2 < Ln) {
            asm volatile("s_wait_dscnt 0x0" ::: "memory");
            ASYNC_BATCH((l & 1), (l + 2))
        }
    }
#undef ASYNC_BATCH
}

extern "C" void kernel_launch(void* const* d_in, const int* in_sizes, int n_in,
                              void* d_out, int out_size, void* d_ws, size_t ws_size,
                              hipStream_t stream) {
    (void)in_sizes; (void)n_in; (void)out_size; (void)d_ws; (void)ws_size;
    // 16 batch rows per 256-thread block (2 per wave32) -> 1024 blocks.
    prop_scan_kernel<<<Bn / 16, 256, 0, stream>>>(
        (const float*)d_in[0],  (const float*)d_in[1],
        (const float*)d_in[2],  (const float*)d_in[3],
        (const float*)d_in[4],  (const float*)d_in[5],
        (const float*)d_in[6],  (const float*)d_in[7],
        (const float*)d_in[8],  (const float*)d_in[9],
        (const float*)d_in[10], (float*)d_out);
}